// EncoderGCN5_75265006895442
// MI455X (gfx1250) — compile-verified
//
#include <hip/hip_runtime.h>

typedef float v2f __attribute__((ext_vector_type(2)));
typedef float v8f __attribute__((ext_vector_type(8)));

// ---------------------------------------------------------------------------
// Dense GEMM H[nrows x DOUT] = X[nrows x DIN] * W[DIN x DOUT], f32, via
// V_WMMA_F32_16X16X4_F32. One wave32 computes one full 16 x DOUT strip of H
// (NT = DOUT/16 accumulator tiles): each A fragment loaded from global is
// reused NT times from registers; B comes from LDS.
// W is staged in LDS in a k-paired layout so each B fragment
// {W[ka][col], W[ka+1][col]} is one contiguous 8-byte ds_load_b64:
//   sW[((k>>1)*DOUT + col)*2 + (k&1)] = W[k*DOUT + col]
// A layout (16x4 f32): lanes 0-15 hold {K=k+0,k+1}, lanes 16-31 {K=k+2,k+3}.
// B layout (4x16 f32): mirrored (row K striped across lanes within a VGPR).
// D layout (16x16 f32, 8 VGPRs): VGPR r -> row r (lanes 0-15), row r+8 (16-31).
// ---------------------------------------------------------------------------
template <int DIN, int DOUT>
__global__ __launch_bounds__(256) void gemm_wmma_f32(
    const float* __restrict__ X, const float* __restrict__ W,
    float* __restrict__ H, int nrows, int mtiles_total) {
  __shared__ float sW[DIN * DOUT];
  for (int i = threadIdx.x; i < DIN * DOUT; i += 256) {
    const int k = i / DOUT;
    const int col = i - k * DOUT;
    sW[(((k >> 1) * DOUT + col) << 1) | (k & 1)] = W[i];
  }
  __syncthreads();

  const int wave = threadIdx.x >> 5;
  const int lane = threadIdx.x & 31;
  const int mtile = blockIdx.x * 8 + wave;
  if (mtile >= mtiles_total) return;

  constexpr int NT = DOUT / 16;

  const int half = lane >> 4;  // 0 for lanes 0-15, 1 for lanes 16-31
  const int l16 = lane & 15;

  int row = mtile * 16 + l16;         // A-matrix row owned by this lane
  if (row >= nrows) row = nrows - 1;  // clamp (nrows is a multiple of 16 here)
  const float* __restrict__ xrow = X + (long)row * DIN;

  v8f acc[NT];
#pragma unroll
  for (int nt = 0; nt < NT; ++nt) acc[nt] = (v8f){};

#pragma unroll
  for (int k = 0; k < DIN; k += 4) {
    const int ka = k + half * 2;  // even
    const v2f a = *(const v2f*)(xrow + ka);
    const int kp = ka >> 1;  // paired-k index for this lane half
#pragma unroll
    for (int nt = 0; nt < NT; ++nt) {
      const int col = nt * 16 + l16;
      const v2f b = *(const v2f*)&sW[(kp * DOUT + col) << 1];
      acc[nt] = __builtin_amdgcn_wmma_f32_16x16x4_f32(false, a, false, b,
                                                      (short)0, acc[nt], false,
                                                      false);
    }
  }

#pragma unroll
  for (int r = 0; r < 8; ++r) {
    const int orow = mtile * 16 + half * 8 + r;
    if (orow < nrows) {
#pragma unroll
      for (int nt = 0; nt < NT; ++nt) {
        H[(long)orow * DOUT + nt * 16 + l16] = acc[nt][r];
      }
    }
  }
}

// ---------------------------------------------------------------------------
// GCN-norm helpers
// ---------------------------------------------------------------------------
__global__ void k_fill(float* __restrict__ p, float v, int n) {
  int i = blockIdx.x * blockDim.x + threadIdx.x;
  if (i < n) p[i] = v;
}

__global__ void k_deg(const int* __restrict__ dst, float* __restrict__ deg,
                      int E) {
  int e = blockIdx.x * blockDim.x + threadIdx.x;
  if (e < E) atomicAdd(&deg[dst[e]], 1.0f);
}

__global__ void k_rsqrt_inplace(float* __restrict__ p, int n) {
  int i = blockIdx.x * blockDim.x + threadIdx.x;
  if (i < n) {
    float d = p[i];
    p[i] = d > 0.0f ? rsqrtf(d) : 0.0f;
  }
}

__global__ void k_edge_norm(const int* __restrict__ src,
                            const int* __restrict__ dst,
                            const float* __restrict__ dinv,
                            float* __restrict__ norm, int E) {
  int e = blockIdx.x * blockDim.x + threadIdx.x;
  if (e < E) norm[e] = dinv[src[e]] * dinv[dst[e]];
}

// out[i,f] = h[i,f] * dinv[i]^2   (self-loop term; also initializes out)
__global__ void k_selfloop_init(const float* __restrict__ h,
                                const float* __restrict__ dinv,
                                float* __restrict__ out, int n_elems,
                                int log2d) {
  int t = blockIdx.x * blockDim.x + threadIdx.x;
  if (t < n_elems) {
    int i = t >> log2d;
    float di = dinv[i];
    out[t] = h[t] * di * di;
  }
}

// out[dst[e],f] += h[src[e],f] * norm[e]
__global__ void k_scatter(const float* __restrict__ h,
                          const int* __restrict__ src,
                          const int* __restrict__ dst,
                          const float* __restrict__ norm,
                          float* __restrict__ out, int total, int log2d) {
  int t = blockIdx.x * blockDim.x + threadIdx.x;
  if (t < total) {
    int e = t >> log2d;
    int f = t & ((1 << log2d) - 1);
    float v = h[((long)src[e] << log2d) + f] * norm[e];
    atomicAdd(&out[((long)dst[e] << log2d) + f], v);
  }
}

__global__ void k_bias_act(float* __restrict__ out,
                           const float* __restrict__ bias, int n_elems,
                           int dmask, int do_relu) {
  int t = blockIdx.x * blockDim.x + threadIdx.x;
  if (t < n_elems) {
    float v = out[t] + bias[t & dmask];
    out[t] = do_relu ? fmaxf(v, 0.0f) : v;
  }
}

// ---------------------------------------------------------------------------
// Host-side orchestration
// ---------------------------------------------------------------------------
static inline int cdiv(int a, int b) { return (a + b - 1) / b; }

static void run_branch(const float* X, const int* ei, int N, int E,
                       void* const* d_in, int param_base, float* out_final,
                       float* dinv, float* norm, float* g, float* aggbuf,
                       hipStream_t stream) {
  const int* src = ei;      // edge_index[0]
  const int* dst = ei + E;  // edge_index[1]
  const int dims[6] = {128, 64, 64, 64, 64, 32};

  // GCN normalization: deg (with self-loop) -> dinv -> per-edge norm
  k_fill<<<cdiv(N, 256), 256, 0, stream>>>(dinv, 1.0f, N);
  k_deg<<<cdiv(E, 256), 256, 0, stream>>>(dst, dinv, E);
  k_rsqrt_inplace<<<cdiv(N, 256), 256, 0, stream>>>(dinv, N);
  k_edge_norm<<<cdiv(E, 256), 256, 0, stream>>>(src, dst, dinv, norm, E);

  const float* cur = X;
  for (int l = 0; l < 5; ++l) {
    const int dout = dims[l + 1];
    const float* W = (const float*)d_in[param_base + 2 * l];
    const float* b = (const float*)d_in[param_base + 2 * l + 1];

    const int mtiles = N / 16;
    const int gblocks = cdiv(mtiles, 8);
    if (l == 0)
      gemm_wmma_f32<128, 64><<<gblocks, 256, 0, stream>>>(cur, W, g, N, mtiles);
    else if (l < 4)
      gemm_wmma_f32<64, 64><<<gblocks, 256, 0, stream>>>(cur, W, g, N, mtiles);
    else
      gemm_wmma_f32<64, 32><<<gblocks, 256, 0, stream>>>(cur, W, g, N, mtiles);

    float* agg = (l == 4) ? out_final : aggbuf;
    const int log2d = (dout == 64) ? 6 : 5;
    const int nel = N * dout;
    k_selfloop_init<<<cdiv(nel, 256), 256, 0, stream>>>(g, dinv, agg, nel,
                                                        log2d);
    const int etotal = E * dout;
    k_scatter<<<cdiv(etotal, 256), 256, 0, stream>>>(g, src, dst, norm, agg,
                                                     etotal, log2d);
    k_bias_act<<<cdiv(nel, 256), 256, 0, stream>>>(agg, b, nel, dout - 1,
                                                   l < 4 ? 1 : 0);
    cur = agg;
  }
}

extern "C" void kernel_launch(void* const* d_in, const int* in_sizes, int n_in,
                              void* d_out, int out_size, void* d_ws,
                              size_t ws_size, hipStream_t stream) {
  const float* x = (const float*)d_in[0];
  const int* x_ei = (const int*)d_in[1];
  const float* y = (const float*)d_in[2];
  const int* y_ei = (const int*)d_in[3];

  const int F_IN = 128;
  const int N = in_sizes[0] / F_IN;  // 50000
  const int E = in_sizes[1] / 2;     // 800000
  const int OUT = 32;

  // Workspace carve (floats): dinv[N] | norm[E] | g[N*64] | agg[N*64]
  float* ws = (float*)d_ws;
  float* dinv = ws;
  float* norm = dinv + N;
  float* g = norm + E;
  float* agg = g + (size_t)N * 64;

  float* out_x = (float*)d_out;            // [N x 32]
  float* out_y = out_x + (size_t)N * OUT;  // [N x 32]

  run_branch(x, x_ei, N, E, d_in, 4, out_x, dinv, norm, g, agg, stream);
  run_branch(y, y_ei, N, E, d_in, 14, out_y, dinv, norm, g, agg, stream);
}